// SelfAttentionLayer_43731357008170
// MI455X (gfx1250) — compile-verified
//
#include <hip/hip_runtime.h>

// ---------------------------------------------------------------------------
// Problem sizes (fixed by the reference)
// ---------------------------------------------------------------------------
#define BATCH 2048
#define NINS  64
#define DIM   768
#define HID   64

// GEMM chunking for the fused kernel
#define KCH    64                 // K-chunk staged in LDS per step
#define NCHUNK (DIM / KCH)        // 12 chunks over K=768
#define LDA    (KCH + 4)          // padded LDS row stride (floats) for A tile
#define LDW    (HID + 4)          // padded LDS row stride for W tile
#define LDH    (HID + 4)          // padded LDS row stride for H tile

// LDS layout (float offsets) -- double-buffered A/W tiles for async pipeline
#define OFF_A0   0
#define OFF_A1   (OFF_A0  + NINS * LDA)    // 4352
#define OFF_WT0  (OFF_A1  + NINS * LDA)    // 8704
#define OFF_WT1  (OFF_WT0 + KCH  * LDW)    // 13056
#define OFF_H    (OFF_WT1 + KCH  * LDW)    // 17408
#define OFF_OPB  (OFF_H   + NINS * LDH)    // 21760 (ownProj + b1 fused)
#define OFF_W2   (OFF_OPB + HID)
#define OFF_ATT  (OFF_W2  + HID)
#define OFF_EX   (OFF_ATT + NINS)
#define OFF_WGT  (OFF_EX  + NINS)
#define SMEM_FLOATS (OFF_WGT + NINS)       // 22080 floats = 88320 B (<320KB WGP LDS)

typedef float v2f __attribute__((ext_vector_type(2)));
typedef float v4f __attribute__((ext_vector_type(4)));
typedef float v8f __attribute__((ext_vector_type(8)));

// ---------------------------------------------------------------------------
// CDNA5 async global->LDS copy (tracked by ASYNCcnt).  16B per lane, LDS
// byte address in a VGPR (%0 = VDST), 64-bit global address in a VGPR pair.
// ---------------------------------------------------------------------------
__device__ __forceinline__ void async_copy_b128(unsigned lds_byte, const float* g) {
    asm volatile("global_load_async_to_lds_b128 %0, %1, off"
                 :: "v"(lds_byte), "v"(g)
                 : "memory");
}
__device__ __forceinline__ void wait_async_le8()  {
    asm volatile("s_wait_asynccnt 0x8" ::: "memory");
}
__device__ __forceinline__ void wait_async_zero() {
    asm volatile("s_wait_asynccnt 0x0" ::: "memory");
}

// ---------------------------------------------------------------------------
// Kernel 1: ownProj[b][j] = own_row(b)[k] * W1[k][j],  k in [0,768)
// own_row(b) = inputs[b, index, :] if index > -1 else claims[b, :]
// ---------------------------------------------------------------------------
__global__ void __launch_bounds__(256)
ownproj_kernel(const float* __restrict__ inputs,
               const float* __restrict__ claims,
               const float* __restrict__ W1,
               const int*   __restrict__ idxp,
               float*       __restrict__ OP)
{
    const int tid = threadIdx.x;
    const int b   = blockIdx.x * 4 + (tid >> 6);
    const int j   = tid & 63;
    const int idx = *idxp;

    const float* __restrict__ src = (idx > -1)
        ? (inputs + (size_t)b * NINS * DIM + (size_t)idx * DIM)
        : (claims + (size_t)b * DIM);

    float acc = 0.0f;
#pragma unroll 4
    for (int k = 0; k < DIM; ++k) {
        acc += src[k] * W1[(size_t)k * HID + j];   // W1 row-major [1536,64], top half
    }
    OP[(size_t)b * HID + j] = acc;
}

// ---------------------------------------------------------------------------
// Per-chunk async stage: A chunk 64xKCH of inputs[b], W chunk KCHx64 of
// W1_bot.  1024 float4 slots each, 256 threads x 4; 8 async ops per thread.
// ---------------------------------------------------------------------------
__device__ __forceinline__ void issue_chunk(const float* __restrict__ inputs,
                                            size_t inbase,
                                            const float* __restrict__ W1,
                                            int kc, int buf, int tid)
{
    const unsigned aBase = (buf ? OFF_A1  : OFF_A0);
    const unsigned wBase = (buf ? OFF_WT1 : OFF_WT0);
#pragma unroll
    for (int it = 0; it < 4; ++it) {
        const int slot = tid + it * 256;
        const int row  = slot >> 4;       // 16 float4 per row
        const int c4   = slot & 15;
        async_copy_b128((aBase + row * LDA + c4 * 4) * 4u,
                        inputs + inbase + (size_t)row * DIM + kc * KCH + c4 * 4);
        async_copy_b128((wBase + row * LDW + c4 * 4) * 4u,
                        W1 + (size_t)(DIM + kc * KCH + row) * HID + c4 * 4);
    }
}

// ---------------------------------------------------------------------------
// Kernel 2: fully fused per-batch attention-MIL pooling.
// 1 workgroup = 1 batch. 256 threads = 8 wave32s.
//   Phase A: H[64,64] = inputs[b] @ W1_bot via V_WMMA_F32_16X16X4_F32,
//            K double-buffered through LDS with async global->LDS copies.
//   Phase B: h = relu(H + ownProj + b1); att[n] = h . W2 + b2; softmax(att).
//   Phase C: out[b,:] = sum_n weight[n] * inputs[b,n,:]  (L2-resident reread).
// ---------------------------------------------------------------------------
__global__ void __launch_bounds__(256)
attnmil_fused_kernel(const float* __restrict__ inputs,
                     const float* __restrict__ W1,
                     const float* __restrict__ b1,
                     const float* __restrict__ W2,
                     const float* __restrict__ b2,
                     const float* __restrict__ OP,
                     float*       __restrict__ out)
{
    extern __shared__ float sm[];

    const int tid  = threadIdx.x;
    const int lane = tid & 31;
    const int wave = tid >> 5;
    const int b    = blockIdx.x;

    const size_t inbase = (size_t)b * NINS * DIM;

    // Stage (ownProj + b1) and W2 once.
    if (tid < HID) {
        sm[OFF_OPB + tid] = OP[(size_t)b * HID + tid] + b1[tid];
        sm[OFF_W2  + tid] = W2[tid];
    }

    // Wave tiling: rows [16*mt, 16*mt+16), cols [32*nh, 32*nh+32)
    const int mt = wave & 3;
    const int nh = wave >> 2;

    v8f acc[2] = {};                       // two 16x16 f32 C tiles

    const int m  = mt * 16 + (lane & 15);  // A-matrix row for this lane
    const int kh = (lane >> 4) << 1;       // K sub-pair select (0 or 2)

    // Prologue: kick off chunk 0.
    issue_chunk(inputs, inbase, W1, 0, 0, tid);

    for (int kc = 0; kc < NCHUNK; ++kc) {
        // Overlap: issue next chunk into the alternate buffer, then wait for
        // this chunk's 8 per-wave copies (async loads complete in order).
        if (kc + 1 < NCHUNK) {
            issue_chunk(inputs, inbase, W1, kc + 1, (kc + 1) & 1, tid);
            wait_async_le8();
        } else {
            wait_async_zero();
        }
        __syncthreads();                   // all waves' chunk-kc data in LDS

        const unsigned aB = ((kc & 1) ? OFF_A1  : OFF_A0);
        const unsigned wB = ((kc & 1) ? OFF_WT1 : OFF_WT0);

        // --- 16 WMMA K-steps over this chunk ---
#pragma unroll 4
        for (int k = 0; k < KCH; k += 4) {
            v2f a = *(const v2f*)&sm[aB + m * LDA + k + kh];
#pragma unroll
            for (int nt = 0; nt < 2; ++nt) {
                const int n = nh * 32 + nt * 16 + (lane & 15);
                v2f bb;
                bb[0] = sm[wB + (k + kh)     * LDW + n];
                bb[1] = sm[wB + (k + kh + 1) * LDW + n];
                acc[nt] = __builtin_amdgcn_wmma_f32_16x16x4_f32(
                    /*neg_a=*/false, a, /*neg_b=*/false, bb,
                    /*c_mod=*/(short)0, acc[nt],
                    /*reuse_a=*/false, /*reuse_b=*/false);
            }
        }
        __syncthreads();                   // buffer free for overwrite
    }

    // --- epilogue: h = relu(acc + ownProj + b1) -> LDS H tile ---
    {
        const int mhi = (lane < 16) ? 0 : 8;
#pragma unroll
        for (int nt = 0; nt < 2; ++nt) {
            const int j = nh * 32 + nt * 16 + (lane & 15);
            const float opb = sm[OFF_OPB + j];
#pragma unroll
            for (int r = 0; r < 8; ++r) {
                const int row = mt * 16 + mhi + r;
                float v = acc[nt][r] + opb;
                sm[OFF_H + row * LDH + j] = v > 0.0f ? v : 0.0f;
            }
        }
    }
    __syncthreads();

    // --- scores: att[n] = h[n,:] . W2 + b2 ---
    if (tid < NINS) {
        float s = 0.0f;
#pragma unroll 8
        for (int j = 0; j < HID; ++j)
            s += sm[OFF_H + tid * LDH + j] * sm[OFF_W2 + j];
        sm[OFF_ATT + tid] = s + b2[0];
    }
    __syncthreads();

    // --- softmax over 64 instances ---
    if (tid < NINS) {
        float mx = sm[OFF_ATT + 0];
#pragma unroll 8
        for (int n = 1; n < NINS; ++n) mx = fmaxf(mx, sm[OFF_ATT + n]);
        sm[OFF_EX + tid] = __expf(sm[OFF_ATT + tid] - mx);
    }
    __syncthreads();
    if (tid < NINS) {
        float sum = 0.0f;
#pragma unroll 8
        for (int n = 0; n < NINS; ++n) sum += sm[OFF_EX + n];
        sm[OFF_WGT + tid] = sm[OFF_EX + tid] / sum;
    }
    __syncthreads();

    // --- weighted pooling: out[b,d] = sum_n w[n] * inputs[b,n,d] (float4) ---
    if (tid < DIM / 4) {
        v4f accp = {};
        const float* __restrict__ base = inputs + inbase;
#pragma unroll 4
        for (int n = 0; n < NINS; ++n) {
            const float w = sm[OFF_WGT + n];
            v4f v = *(const v4f*)(base + (size_t)n * DIM + tid * 4);
            accp += w * v;
        }
        *(v4f*)(out + (size_t)b * DIM + tid * 4) = accp;
    }
}

// ---------------------------------------------------------------------------
// Host launcher
// ---------------------------------------------------------------------------
extern "C" void kernel_launch(void* const* d_in, const int* in_sizes, int n_in,
                              void* d_out, int out_size, void* d_ws, size_t ws_size,
                              hipStream_t stream) {
    (void)in_sizes; (void)n_in; (void)out_size; (void)ws_size;

    const float* inputs = (const float*)d_in[0];   // [2048, 64, 768]
    const float* claims = (const float*)d_in[1];   // [2048, 768]
    const float* W1     = (const float*)d_in[2];   // [1536, 64]
    const float* b1     = (const float*)d_in[3];   // [64]
    const float* W2     = (const float*)d_in[4];   // [64, 1]
    const float* b2     = (const float*)d_in[5];   // [1]
    const int*   idx    = (const int*)d_in[6];     // scalar

    float* OP  = (float*)d_ws;                     // [2048, 64] ownProj scratch
    float* out = (float*)d_out;                    // [2048, 768]

    ownproj_kernel<<<BATCH / 4, 256, 0, stream>>>(inputs, claims, W1, idx, OP);

    attnmil_fused_kernel<<<BATCH, 256, SMEM_FLOATS * sizeof(float), stream>>>(
        inputs, W1, b1, W2, b2, OP, out);
}